// Loss_tgm_29996051595507
// MI455X (gfx1250) — compile-verified
//
#include <hip/hip_runtime.h>

// Problem constants (B=4, N=32, H=W=518)
#define HW      268324          // 518*518
#define HWQ     67081           // HW/4 : float4 elements per frame
#define NPAIR   31              // N-1
#define NBATCH  4
#define PAIRS   (NBATCH * NPAIR)   // 124
#define BPP     16              // blocks per pair
#define THREADS 256             // 8 wave32s
#define STRIDE  (BPP * THREADS) // 4096

typedef __attribute__((ext_vector_type(2))) float v2f;
typedef __attribute__((ext_vector_type(8))) float v8f;

// Exact f32 wave32 sum using V_WMMA_F32_16X16X4_F32 as a K-reducer:
//   A = ones (16x4), B: VGPR0 = lane value (K=0 lanes 0-15, K=1 lanes 16-31),
//   VGPR1 = 0 (K=2,3). D[m][n] = B[0][n] + B[1][n] = v[n] + v[n+16],
//   replicated in every lane's D VGPR0 (lane L holds column L%16).
// Four shfl_xor steps then fold the 16 column sums. All adds are f32-exact.
__device__ __forceinline__ float wave_sum_wmma(float v) {
    v2f a;  a[0] = 1.0f; a[1] = 1.0f;
    v2f bm; bm[0] = v;   bm[1] = 0.0f;
    v8f cz = {};
    v8f d = __builtin_amdgcn_wmma_f32_16x16x4_f32(
        /*neg_a=*/false, a, /*neg_b=*/false, bm,
        /*c_mod=*/(short)0, cz, /*reuse_a=*/false, /*reuse_b=*/false);
    float x = d[0];                 // x(lane) = v[lane%16] + v[(lane%16)+16]
    x += __shfl_xor(x, 1, 32);
    x += __shfl_xor(x, 2, 32);
    x += __shfl_xor(x, 4, 32);
    x += __shfl_xor(x, 8, 32);
    return x;                       // full 32-lane sum, in every lane
}

__global__ void tgm_partial_kernel(const float* __restrict__ pred,
                                   const float* __restrict__ y,
                                   const unsigned char* __restrict__ msk,
                                   float2* __restrict__ partials) {
    const int pair  = blockIdx.x / BPP;
    const int chunk = blockIdx.x % BPP;
    const int b = pair / NPAIR;
    const int n = pair % NPAIR;

    const size_t f0 = (size_t)(b * 32 + n) * (size_t)HWQ;  // frame n  (float4 units)
    const size_t f1 = f0 + (size_t)HWQ;                    // frame n+1

    const float4* __restrict__ p0 = (const float4*)pred + f0;
    const float4* __restrict__ p1 = (const float4*)pred + f1;
    const float4* __restrict__ g0 = (const float4*)y    + f0;
    const float4* __restrict__ g1 = (const float4*)y    + f1;
    const unsigned int* __restrict__ m0 = (const unsigned int*)msk + f0;
    const unsigned int* __restrict__ m1 = (const unsigned int*)msk + f1;

    float s = 0.0f;   // masked sum of ||dpred| - |dy||
    float c = 0.0f;   // static count (exact in f32: <= ~68 per lane)

    for (int i = chunk * THREADS + (int)threadIdx.x; i < HWQ; i += STRIDE) {
        const float4 a0 = p0[i];
        const float4 a1 = p1[i];
        const float4 b0 = g0[i];
        const float4 b1 = g1[i];
        const unsigned int vv = m0[i] & m1[i];  // bool bytes are 0/1 -> byte-wise AND

        {
            float gd = fabsf(b1.x - b0.x);
            bool  st = (gd < 0.05f) && ((vv & 0x000000FFu) != 0u);
            s += st ? fabsf(fabsf(a1.x - a0.x) - gd) : 0.0f;
            c += st ? 1.0f : 0.0f;
        }
        {
            float gd = fabsf(b1.y - b0.y);
            bool  st = (gd < 0.05f) && ((vv & 0x0000FF00u) != 0u);
            s += st ? fabsf(fabsf(a1.y - a0.y) - gd) : 0.0f;
            c += st ? 1.0f : 0.0f;
        }
        {
            float gd = fabsf(b1.z - b0.z);
            bool  st = (gd < 0.05f) && ((vv & 0x00FF0000u) != 0u);
            s += st ? fabsf(fabsf(a1.z - a0.z) - gd) : 0.0f;
            c += st ? 1.0f : 0.0f;
        }
        {
            float gd = fabsf(b1.w - b0.w);
            bool  st = (gd < 0.05f) && ((vv & 0xFF000000u) != 0u);
            s += st ? fabsf(fabsf(a1.w - a0.w) - gd) : 0.0f;
            c += st ? 1.0f : 0.0f;
        }
    }

    // Wave reduction on the matrix pipe (EXEC is all-1s here: post-loop reconvergence).
    const float ws = wave_sum_wmma(s);
    const float wc = wave_sum_wmma(c);

    __shared__ float sh_s[THREADS / 32];
    __shared__ float sh_c[THREADS / 32];
    const int lane = threadIdx.x & 31;
    const int wid  = threadIdx.x >> 5;
    if (lane == 0) { sh_s[wid] = ws; sh_c[wid] = wc; }
    __syncthreads();

    if (threadIdx.x == 0) {
        float S = 0.0f, C = 0.0f;
#pragma unroll
        for (int k = 0; k < THREADS / 32; ++k) { S += sh_s[k]; C += sh_c[k]; }
        partials[pair * BPP + chunk] = make_float2(S, C);
    }
}

__global__ void tgm_final_kernel(const float2* __restrict__ partials,
                                 float* __restrict__ out) {
    __shared__ float red[128];
    const int t = threadIdx.x;      // 128 threads

    float tgm = 0.0f;
    if (t < PAIRS) {
        float s = 0.0f, c = 0.0f;
#pragma unroll
        for (int k = 0; k < BPP; ++k) {
            const float2 v = partials[t * BPP + k];
            s += v.x;
            c += v.y;
        }
        tgm = (c > 0.0f) ? (s / c) : 0.0f;
    }
    red[t] = tgm;
    __syncthreads();

    for (int w = 64; w > 0; w >>= 1) {
        if (t < w) red[t] += red[t + w];
        __syncthreads();
    }
    if (t == 0) out[0] = red[0] / (float)PAIRS;
}

extern "C" void kernel_launch(void* const* d_in, const int* in_sizes, int n_in,
                              void* d_out, int out_size, void* d_ws, size_t ws_size,
                              hipStream_t stream) {
    (void)in_sizes; (void)n_in; (void)out_size; (void)ws_size;

    const float*         pred = (const float*)d_in[0];
    const float*         y    = (const float*)d_in[1];
    const unsigned char* msk  = (const unsigned char*)d_in[2];
    float*               out  = (float*)d_out;
    float2*              part = (float2*)d_ws;   // 124*16 float2 = 15,872 bytes

    tgm_partial_kernel<<<PAIRS * BPP, THREADS, 0, stream>>>(pred, y, msk, part);
    tgm_final_kernel<<<1, 128, 0, stream>>>(part, out);
}